// MutationMLMLossModel_59167469469987
// MI455X (gfx1250) — compile-verified
//
#include <hip/hip_runtime.h>
#include <stdint.h>

#define B_DIM 256
#define S_DIM 4096
#define V_DIM 30
#define BS (B_DIM * S_DIM)                      // 1,048,576 rows
#define ROW_BYTES (V_DIM * 4)                   // 120 B per row
#define NUM_BLOCKS 2048
#define THREADS 128
#define ROWS_PER_BLOCK (BS / NUM_BLOCKS)        // 512
#define TILE_ROWS 256
#define TILES (ROWS_PER_BLOCK / TILE_ROWS)      // 2
#define TILE_BYTES (TILE_ROWS * ROW_BYTES)      // 30720 B
#define TILE_CHUNKS (TILE_BYTES / 16)           // 1920 x b128 chunks
// 1920 / 128 threads = 15 async ops per thread per tile (uniform, per-wave ASYNCcnt += 15)

typedef __attribute__((ext_vector_type(2))) float v2f;
typedef __attribute__((ext_vector_type(8))) float v8f;

// CDNA5 async global->LDS copy, 16B per lane (ASYNCcnt-tracked).
// saddr (GVS) form: mem_addr = SGPR64 + VGPR32 + ioffset; vdst = LDS byte addr.
__device__ __forceinline__ void async_copy_b128(uint32_t lds_off, uint64_t gbase, uint32_t goff) {
    asm volatile("global_load_async_to_lds_b128 %0, %1, %2"
                 :: "v"(lds_off), "v"(goff), "s"(gbase)
                 : "memory");
}
__device__ __forceinline__ void wait_asynccnt_15() { asm volatile("s_wait_asynccnt 15" ::: "memory"); }
__device__ __forceinline__ void wait_asynccnt_0()  { asm volatile("s_wait_asynccnt 0"  ::: "memory"); }

__global__ void __launch_bounds__(THREADS)
mlm_kernel(const float* __restrict__ mlm, const int* __restrict__ tgt,
           float* __restrict__ out_pred, float* __restrict__ ws) {
    __shared__ float lbuf[2][TILE_ROWS * V_DIM];   // 2 x 30720 B double buffer
    __shared__ float red[THREADS];

    const int tid = threadIdx.x;
    const int rowBase = blockIdx.x * ROWS_PER_BLOCK;
    const uint64_t gbase = (uint64_t)(uintptr_t)mlm;
    const uint32_t blockByteBase = (uint32_t)rowBase * ROW_BYTES;   // total < 2^27, fits u32

    // Prefetch tile 0 -> buffer 0 (15 async b128 ops per thread)
    {
        const uint32_t lb = (uint32_t)(uintptr_t)&lbuf[0][0];
        for (int c = tid; c < TILE_CHUNKS; c += THREADS)
            async_copy_b128(lb + (uint32_t)c * 16u, gbase, blockByteBase + (uint32_t)c * 16u);
    }

    float lsum = 0.0f;
    for (int t = 0; t < TILES; ++t) {
        const int bi = t & 1;
        if (t + 1 < TILES) {
            // Prefetch next tile into the other buffer, then wait for the
            // oldest 15 async ops (tile t) — async loads complete in order.
            const uint32_t lb = (uint32_t)(uintptr_t)&lbuf[(t + 1) & 1][0];
            const uint32_t gB = blockByteBase + (uint32_t)(t + 1) * TILE_BYTES;
            for (int c = tid; c < TILE_CHUNKS; c += THREADS)
                async_copy_b128(lb + (uint32_t)c * 16u, gbase, gB + (uint32_t)c * 16u);
            wait_asynccnt_15();
        } else {
            wait_asynccnt_0();
        }
        __syncthreads();   // tile t resident for all waves

        // Each thread handles rows tid and tid+128 of this tile.
        // LDS bank = (30*lane + v) mod 64 -> conflict-free across a wave32.
        for (int lr = tid; lr < TILE_ROWS; lr += THREADS) {
            const float* row = &lbuf[bi][lr * V_DIM];
            float m = row[0];
            int am = 0;
            #pragma unroll
            for (int v = 1; v < V_DIM; ++v) {
                float x = row[v];
                if (x > m) { m = x; am = v; }   // strict > keeps first max (jnp.argmax)
            }
            float s = 0.0f;
            #pragma unroll
            for (int v = 0; v < V_DIM; ++v) s += expf(row[v] - m);
            const int gRow = rowBase + t * TILE_ROWS + lr;
            const int tg = tgt[gRow];
            const float lse = logf(s) + m;
            if (tg != 0) lsum += lse - row[tg];   // masked NLL (PAD_INDEX == 0)
            out_pred[gRow] = (float)am;           // coalesced store
        }
        __syncthreads();   // buffer reuse fence before next prefetch overwrites it
    }

    // Deterministic block partial sum -> d_ws[blockIdx.x]
    red[tid] = lsum;
    __syncthreads();
    for (int s = THREADS / 2; s > 0; s >>= 1) {
        if (tid < s) red[tid] += red[tid + s];
        __syncthreads();
    }
    if (tid == 0) ws[blockIdx.x] = red[0];
}

__global__ void __launch_bounds__(256)
finalize_kernel(const float* __restrict__ nsp, const float* __restrict__ ws,
                float* __restrict__ out) {
    __shared__ float s256[256];
    const int tid = threadIdx.x;

    // NSP argmax over 2 logits (ties -> first index, matches jnp.argmax)
    {
        const float v0 = nsp[2 * tid];
        const float v1 = nsp[2 * tid + 1];
        out[BS + tid] = (v1 > v0) ? 1.0f : 0.0f;
    }

    // Gather 2048 block partials in fixed order -> 256 sums
    float p = 0.0f;
    #pragma unroll
    for (int k = 0; k < NUM_BLOCKS / 256; ++k) p += ws[tid + k * 256];
    s256[tid] = p;
    __syncthreads();

    // Fold 256 -> 64 values (A matrix, 16x4 f32), then one f32 WMMA with
    // B = ones computes row sums; sum of the 8 D-VGPRs per lane gives
    // S_low (lanes 0..15) / S_high (lanes 16..31); xor-shuffle adds them.
    // All waves compute identically with full EXEC (WMMA requirement).
    const int lane = tid & 31;
    v2f a;
    a.x = s256[2 * lane]     + s256[2 * lane + 64] + s256[2 * lane + 128] + s256[2 * lane + 192];
    a.y = s256[2 * lane + 1] + s256[2 * lane + 65] + s256[2 * lane + 129] + s256[2 * lane + 193];
#if defined(__has_builtin) && __has_builtin(__builtin_amdgcn_wmma_f32_16x16x4_f32)
    v2f b; b.x = 1.0f; b.y = 1.0f;
    v8f c = {};
    c = __builtin_amdgcn_wmma_f32_16x16x4_f32(
            /*neg_a=*/false, a, /*neg_b=*/false, b,
            /*c_mod=*/(short)0, c, /*reuse_a=*/false, /*reuse_b=*/false);
    float s8 = c[0] + c[1] + c[2] + c[3] + c[4] + c[5] + c[6] + c[7];
    const float total = s8 + __shfl_xor(s8, 16, 32);
    if (tid == 0) out[BS + B_DIM] = total / (float)BS;
#else
    // Fallback: deterministic serial reduce by thread 0
    if (tid == 0) {
        float tot = 0.0f;
        for (int i = 0; i < 256; ++i) tot += s256[i];
        tot += (a.x - a.x); // keep a referenced
        out[BS + B_DIM] = tot / (float)BS;
    }
#endif
}

extern "C" void kernel_launch(void* const* d_in, const int* in_sizes, int n_in,
                              void* d_out, int out_size, void* d_ws, size_t ws_size,
                              hipStream_t stream) {
    const float* mlm = (const float*)d_in[0];   // [256,4096,30] f32
    const float* nsp = (const float*)d_in[1];   // [256,2] f32
    // d_in[2] mutation_matrix: mathematically cancels; unused
    const int*   tgt = (const int*)d_in[3];     // [256,4096] i32
    // d_in[4] is_nexts: unused by the reference's returned values
    float* out = (float*)d_out;                 // [BS preds | B nsp preds | loss]
    float* ws  = (float*)d_ws;                  // NUM_BLOCKS f32 partials (8 KB)

    mlm_kernel<<<NUM_BLOCKS, THREADS, 0, stream>>>(mlm, tgt, out, ws);
    finalize_kernel<<<1, 256, 0, stream>>>(nsp, ws, out);
}